// RNN_Main_Model_8392366096778
// MI455X (gfx1250) — compile-verified
//
#include <hip/hip_runtime.h>

typedef __attribute__((ext_vector_type(2))) float v2f;
typedef __attribute__((ext_vector_type(8))) float v8f;

#define BB 128
#define TT 2048
#define FF 124
#define HH 3
#define OO 6
#define PH 4   // padded hidden stride in workspace (float4-aligned)

// ---------------------------------------------------------------------------
// Kernel 1: pre[t, b, h] = x[b, t, :] . w_ih0[h, :] + b_ih0[h] + b_hh0[h]
// One wave32 per (b, 16-timestep tile). 31 chained V_WMMA_F32_16X16X4_F32.
//
// A (16x4 f32) per-lane layout (ISA 7.12.2): lanes 0-15 = rows M=0..15 with
// K = {0,1} in VGPR0/1; lanes 16-31 = same rows with K = {2,3}.
// => lane loads 2 contiguous floats at xrow[4*kk + 2*hi].
// B (4x16 f32): VGPR0/1 hold rows K={2*hi, 2*hi+1}, col N = lane&15.
// N >= 3 columns are zero (clamped load + select keeps EXEC uniform).
// ---------------------------------------------------------------------------
__global__ __launch_bounds__(32) void rnn_inproj_wmma(
    const float* __restrict__ x,
    const float* __restrict__ w_ih0,
    const float* __restrict__ b_ih0,
    const float* __restrict__ b_hh0,
    float* __restrict__ pre)
{
    const int lane = threadIdx.x;
    const int row  = lane & 15;   // M (time row) for A, N (hidden col) for B
    const int hi   = lane >> 4;   // K half-select
    const int t0   = blockIdx.x * 16;
    const int b    = blockIdx.y;

    const float* xrow = x + ((size_t)b * TT + (size_t)(t0 + row)) * FF;

    const int   n     = row;
    const int   nclmp = (n < HH) ? n : (HH - 1);
    const float nmask = (n < HH) ? 1.0f : 0.0f;
    const float* wrow = w_ih0 + (size_t)nclmp * FF;

    v8f acc = {};
#pragma unroll 4
    for (int kk = 0; kk < FF / 4; ++kk) {
        const int k0 = kk * 4 + 2 * hi;
        v2f a = *(const v2f*)(xrow + k0);       // 8B contiguous per lane
        v2f bm;
        bm.x = wrow[k0]     * nmask;
        bm.y = wrow[k0 + 1] * nmask;
        acc = __builtin_amdgcn_wmma_f32_16x16x4_f32(
            /*neg_a=*/false, a, /*neg_b=*/false, bm,
            /*c_mod=*/(short)0, acc, /*reuse_a=*/false, /*reuse_b=*/false);
    }

    // D layout: VGPR r -> (M = r + 8*hi, N = lane&15)
    if (n < HH) {
        const float bias = b_ih0[n] + b_hh0[n];
#pragma unroll
        for (int r = 0; r < 8; ++r) {
            const int t = t0 + r + 8 * hi;
            pre[((size_t)t * BB + b) * PH + n] = acc[r] + bias;
        }
    }
}

// ---------------------------------------------------------------------------
// Kernel 2: fused two-layer tanh scan + SELU + linear head.
// One lane per batch element; all weights/state in registers.
// ---------------------------------------------------------------------------
__device__ __forceinline__ float fast_tanh(float x) {
    float r;
    // CDNA5 hardware tanh (TRANS op). Trailing v_nop satisfies the
    // 1-independent-op TRANS hazard rule (compiler can't see through asm).
    asm volatile("v_tanh_f32 %0, %1\n\tv_nop" : "=v"(r) : "v"(x));
    return r;
}

__device__ __forceinline__ float selu_f(float x) {
    const float kAlpha = 1.6732632423543772f;
    const float kScale = 1.0507009873554805f;
    const float e = __builtin_amdgcn_exp2f(x * 1.4426950408889634f); // v_exp_f32
    return (x > 0.0f) ? kScale * x : kScale * kAlpha * (e - 1.0f);
}

__global__ __launch_bounds__(32) void rnn_scan(
    const float* __restrict__ pre,
    const float* __restrict__ hx,
    const float* __restrict__ w_hh0,
    const float* __restrict__ w_ih1,
    const float* __restrict__ w_hh1,
    const float* __restrict__ b_ih1,
    const float* __restrict__ b_hh1,
    const float* __restrict__ W_out,
    const float* __restrict__ b_out,
    float* __restrict__ out)
{
    const int b = blockIdx.x * 32 + threadIdx.x;

    float A0[3][3], Wi[3][3], Wh[3][3], c1[3];
#pragma unroll
    for (int i = 0; i < 3; ++i) {
#pragma unroll
        for (int j = 0; j < 3; ++j) {
            A0[i][j] = w_hh0[i * 3 + j];   // uniform -> s_load broadcast
            Wi[i][j] = w_ih1[i * 3 + j];
            Wh[i][j] = w_hh1[i * 3 + j];
        }
        c1[i] = b_ih1[i] + b_hh1[i];
    }
    float Wo[6][3], bo[6];
#pragma unroll
    for (int o = 0; o < 6; ++o) {
        bo[o] = b_out[o];
#pragma unroll
        for (int j = 0; j < 3; ++j) Wo[o][j] = W_out[o * 3 + j];
    }

    float h1[3], h2[3];
#pragma unroll
    for (int i = 0; i < 3; ++i) {
        h1[i] = hx[(0 * BB + b) * HH + i];
        h2[i] = hx[(1 * BB + b) * HH + i];
    }

    for (int t = 0; t < TT; ++t) {
        const float4 pv = *(const float4*)(pre + ((size_t)t * BB + b) * PH);
        const float p[3] = {pv.x, pv.y, pv.z};

        // layer 0: h1 = tanh(p + h1 @ w_hh0^T)
        float n0[3];
#pragma unroll
        for (int i = 0; i < 3; ++i) {
            float u = p[i];
#pragma unroll
            for (int j = 0; j < 3; ++j) u = fmaf(A0[i][j], h1[j], u);
            n0[i] = fast_tanh(u);
        }
#pragma unroll
        for (int i = 0; i < 3; ++i) h1[i] = n0[i];

        // layer 1: h2 = tanh(w_ih1 @ h1 + c1 + h2 @ w_hh1^T)
        float n1[3];
#pragma unroll
        for (int i = 0; i < 3; ++i) {
            float v = c1[i];
#pragma unroll
            for (int j = 0; j < 3; ++j) v = fmaf(Wi[i][j], h1[j], v);
#pragma unroll
            for (int j = 0; j < 3; ++j) v = fmaf(Wh[i][j], h2[j], v);
            n1[i] = fast_tanh(v);
        }
#pragma unroll
        for (int i = 0; i < 3; ++i) h2[i] = n1[i];

        // SELU + linear head
        float y[3];
#pragma unroll
        for (int i = 0; i < 3; ++i) y[i] = selu_f(h2[i]);

        float* op = out + ((size_t)b * TT + t) * OO;
#pragma unroll
        for (int o = 0; o < 6; ++o) {
            float s = bo[o];
#pragma unroll
            for (int j = 0; j < 3; ++j) s = fmaf(Wo[o][j], y[j], s);
            op[o] = s;
        }
    }
}

// ---------------------------------------------------------------------------
extern "C" void kernel_launch(void* const* d_in, const int* in_sizes, int n_in,
                              void* d_out, int out_size, void* d_ws, size_t ws_size,
                              hipStream_t stream)
{
    const float* x     = (const float*)d_in[0];
    const float* hx    = (const float*)d_in[1];
    const float* w_ih0 = (const float*)d_in[2];
    const float* w_hh0 = (const float*)d_in[3];
    const float* b_ih0 = (const float*)d_in[4];
    const float* b_hh0 = (const float*)d_in[5];
    const float* w_ih1 = (const float*)d_in[6];
    const float* w_hh1 = (const float*)d_in[7];
    const float* b_ih1 = (const float*)d_in[8];
    const float* b_hh1 = (const float*)d_in[9];
    const float* W_out = (const float*)d_in[10];
    const float* b_out = (const float*)d_in[11];

    float* out = (float*)d_out;
    float* pre = (float*)d_ws;   // needs TT*BB*PH*4 = 4 MiB of scratch

    dim3 g1(TT / 16, BB);        // 128 x 128 waves
    rnn_inproj_wmma<<<g1, 32, 0, stream>>>(x, w_ih0, b_ih0, b_hh0, pre);

    rnn_scan<<<BB / 32, 32, 0, stream>>>(pre, hx, w_hh0, w_ih1, w_hh1,
                                         b_ih1, b_hh1, W_out, b_out, out);
}